// WassersteinLog_67525475827937
// MI455X (gfx1250) — compile-verified
//
#include <hip/hip_runtime.h>
#include <stdint.h>

#define TPB 256
#define EPSF 1e-8f

// ---------- CDNA5 async data movers (inline asm, ASYNCcnt-tracked) ----------

__device__ __forceinline__ unsigned lds_off_u32(const void* p) {
  // gfx1250 flat LDS aperture keeps the LDS byte offset in addr[31:0]
  // (ISA 10.2).
  return (unsigned)(uintptr_t)p;
}

__device__ __forceinline__ void async_load_b128(unsigned lds_addr, const float* gaddr) {
  asm volatile("global_load_async_to_lds_b128 %0, %1, off"
               :: "v"(lds_addr), "v"(gaddr) : "memory");
}
__device__ __forceinline__ void async_load_b32(unsigned lds_addr, const float* gaddr) {
  asm volatile("global_load_async_to_lds_b32 %0, %1, off"
               :: "v"(lds_addr), "v"(gaddr) : "memory");
}
__device__ __forceinline__ void async_store_b128(const float* gaddr, unsigned lds_addr) {
  asm volatile("global_store_async_from_lds_b128 %0, %1, off"
               :: "v"(gaddr), "v"(lds_addr) : "memory");
}
__device__ __forceinline__ void async_store_b32(const float* gaddr, unsigned lds_addr) {
  asm volatile("global_store_async_from_lds_b32 %0, %1, off"
               :: "v"(gaddr), "v"(lds_addr) : "memory");
}
__device__ __forceinline__ void wait_async0() {
  asm volatile("s_wait_asynccnt 0" ::: "memory");
}

// ---------- fast scalar ops (single-instruction v_rcp/v_sqrt, ~1 ulp) ----------

__device__ __forceinline__ float frcp(float x)  { return __builtin_amdgcn_rcpf(x); }
__device__ __forceinline__ float fsqrt(float x) { return __builtin_amdgcn_sqrtf(x); }

// ---------- small-matrix helpers ----------

// row-major 3x3 multiply C = A*B
__device__ __forceinline__ void mm3(const float* A, const float* B, float* C) {
#pragma unroll
  for (int i = 0; i < 3; ++i) {
#pragma unroll
    for (int j = 0; j < 3; ++j) {
      C[i * 3 + j] = fmaf(A[i * 3 + 0], B[0 + j],
                     fmaf(A[i * 3 + 1], B[3 + j],
                          A[i * 3 + 2] * B[6 + j]));
    }
  }
}

// expand symmetric 6 (s00,s01,s02,s11,s12,s22) -> full 9 row-major
__device__ __forceinline__ void sym_expand(const float* s, float* f) {
  f[0] = s[0]; f[1] = s[1]; f[2] = s[2];
  f[3] = s[1]; f[4] = s[3]; f[5] = s[4];
  f[6] = s[2]; f[7] = s[4]; f[8] = s[5];
}

// Analytic symmetric-sqrt of a 3x3 SPD matrix (6-entry storage).
// Eigenvalues via the closed-form trig method; sqrt via the Newton
// interpolating polynomial with cancellation-free divided differences:
//   f01  = 1/(s0+s1),  f012 = -1/((s0+s1)(s1+s2)(s0+s2)),  s_i = sqrt(l_i)
// Matches the reference's Q diag((L^2)^0.25) Q^T == Q diag(sqrt(|L|)) Q^T
// for SPD inputs (eigenvalues clamped at 0).
__device__ __forceinline__ void sym_sqrt(const float* A6, float* S6) {
  const float a00 = A6[0], a01 = A6[1], a02 = A6[2];
  const float a11 = A6[3], a12 = A6[4], a22 = A6[5];

  // --- eigenvalues ---
  const float m = (a00 + a11 + a22) * (1.0f / 3.0f);
  const float b00 = a00 - m, b11 = a11 - m, b22 = a22 - m;
  const float off2 = a01 * a01 + a02 * a02 + a12 * a12;
  const float p2 = b00 * b00 + b11 * b11 + b22 * b22 + 2.0f * off2;
  const float p = fsqrt(p2 * (1.0f / 6.0f));
  const float invp = frcp(fmaxf(p, 1e-30f));
  const float c00 = b00 * invp, c11 = b11 * invp, c22 = b22 * invp;
  const float c01 = a01 * invp, c02 = a02 * invp, c12 = a12 * invp;
  float r = 0.5f * (c00 * (c11 * c22 - c12 * c12)
                  - c01 * (c01 * c22 - c12 * c02)
                  + c02 * (c01 * c12 - c11 * c02));
  r = fminf(1.0f, fmaxf(-1.0f, r));
  const float phi = acosf(r) * (1.0f / 3.0f);          // phi in [0, pi/3]
  const float cp = cosf(phi);
  const float sp = fsqrt(fmaxf(0.0f, 1.0f - cp * cp)); // sin(phi) >= 0 here
  const float l0 = m + 2.0f * p * cp;                                 // largest
  const float l2 = m + p * fmaf(-1.7320508075688772f, sp, -cp);       // smallest
  const float l1 = 3.0f * m - l0 - l2;

  // --- sqrt coefficients (cancellation-free) ---
  const float s0 = fsqrt(fmaxf(l0, 0.0f));
  const float s1 = fsqrt(fmaxf(l1, 0.0f));
  const float s2 = fsqrt(fmaxf(l2, 0.0f));
  const float f01 = frcp(fmaxf(s0 + s1, 1e-20f));
  const float f012 = -f01 * frcp(fmaxf((s1 + s2) * (s0 + s2), 1e-30f));
  const float gamma = f012;
  const float beta = f01 - f012 * (l0 + l1);
  const float alpha = fmaf(f012 * l0, l1, fmaf(-f01, l0, s0));

  // --- A^2 (symmetric, 6 entries) ---
  const float q00 = a00 * a00 + a01 * a01 + a02 * a02;
  const float q01 = a00 * a01 + a01 * a11 + a02 * a12;
  const float q02 = a00 * a02 + a01 * a12 + a02 * a22;
  const float q11 = a01 * a01 + a11 * a11 + a12 * a12;
  const float q12 = a01 * a02 + a11 * a12 + a12 * a22;
  const float q22 = a02 * a02 + a12 * a12 + a22 * a22;

  S6[0] = fmaf(gamma, q00, fmaf(beta, a00, alpha));
  S6[1] = fmaf(gamma, q01, beta * a01);
  S6[2] = fmaf(gamma, q02, beta * a02);
  S6[3] = fmaf(gamma, q11, fmaf(beta, a11, alpha));
  S6[4] = fmaf(gamma, q12, beta * a12);
  S6[5] = fmaf(gamma, q22, fmaf(beta, a22, alpha));
}

// ---------- kernel ----------

__global__ void __launch_bounds__(TPB)
wasserstein_log_kernel(const float* __restrict__ miu1,
                       const float* __restrict__ miu2,
                       const float* __restrict__ cov1,
                       const float* __restrict__ cov2,
                       float* __restrict__ out_miu,
                       float* __restrict__ out_cov,
                       int B) {
  __shared__ float lds_c1[TPB * 9];   // cov1 tile in, reused as cov_velocity out
  __shared__ float lds_c2[TPB * 9];   // cov2 tile in

  const int tid = threadIdx.x;
  const int base_e = blockIdx.x * TPB;   // first element handled by this block
  const int base_f = base_e * 9;         // first cov float of this block
  const int nfTot = B * 9;

  const unsigned ldsBase1 = lds_off_u32(lds_c1);
  const unsigned ldsBase2 = lds_off_u32(lds_c2);

  // ---- async coalesced global->LDS staging: 2 x 2304 floats per block ----
#pragma unroll
  for (int k = 0; k < 3; ++k) {
    int i4 = k * TPB + tid;                       // float4 index in tile [0,576)
    if (i4 < (TPB * 9) / 4) {
      int gf = base_f + i4 * 4;                   // global float index
      if (gf + 4 <= nfTot) {
        async_load_b128(ldsBase1 + (unsigned)i4 * 16u, cov1 + gf);
        async_load_b128(ldsBase2 + (unsigned)i4 * 16u, cov2 + gf);
      } else if (gf < nfTot) {
#pragma unroll
        for (int j = 0; j < 4; ++j) {
          if (gf + j < nfTot) {
            async_load_b32(ldsBase1 + (unsigned)(i4 * 16 + j * 4), cov1 + gf + j);
            async_load_b32(ldsBase2 + (unsigned)(i4 * 16 + j * 4), cov2 + gf + j);
          }
        }
      }
    }
  }

  // ---- miu diff: elementwise, fully coalesced, overlaps async loads ----
  {
    const int miuTot = B * 3;
    const int base_m = base_e * 3;
#pragma unroll
    for (int k = 0; k < 3; ++k) {
      int idx = base_m + k * TPB + tid;
      if (idx < miuTot) out_miu[idx] = miu2[idx] - miu1[idx];
    }
  }

  wait_async0();        // each wave drains its own ASYNCcnt before the barrier
  __syncthreads();

  const int b = base_e + tid;
  if (b < B) {
    float c1[9], c2[9];
#pragma unroll
    for (int i = 0; i < 9; ++i) {       // stride-9 words: conflict-free on 64 banks
      c1[i] = lds_c1[tid * 9 + i];
      c2[i] = lds_c2[tid * 9 + i];
    }

    // A = cov_1 + eps*I  (symmetrize off-diagonals for the eigensolver)
    float A6[6];
    A6[0] = c1[0] + EPSF;
    A6[1] = 0.5f * (c1[1] + c1[3]);
    A6[2] = 0.5f * (c1[2] + c1[6]);
    A6[3] = c1[4] + EPSF;
    A6[4] = 0.5f * (c1[5] + c1[7]);
    A6[5] = c1[8] + EPSF;

    // A_sqrt
    float S6[6];
    sym_sqrt(A6, S6);

    // A_sqrt_inv = inv(A_sqrt + eps*I) via adjugate (symmetric)
    const float m00 = S6[0] + EPSF, m01 = S6[1], m02 = S6[2];
    const float m11 = S6[3] + EPSF, m12 = S6[4], m22 = S6[5] + EPSF;
    const float co00 = m11 * m22 - m12 * m12;
    const float co01 = m02 * m12 - m01 * m22;
    const float co02 = m01 * m12 - m02 * m11;
    const float det = m00 * co00 + m01 * co01 + m02 * co02;
    const float rdet = frcp(det);
    float I6[6];
    I6[0] = co00 * rdet;
    I6[1] = co01 * rdet;
    I6[2] = co02 * rdet;
    I6[3] = (m00 * m22 - m02 * m02) * rdet;
    I6[4] = (m01 * m02 - m00 * m12) * rdet;
    I6[5] = (m00 * m11 - m01 * m01) * rdet;

    // C = A_sqrt * cov_2 * A_sqrt^T   (A_sqrt symmetric); only upper-6 needed
    float Sf[9], T[9];
    sym_expand(S6, Sf);
    mm3(Sf, c2, T);
    float C6[6];
    C6[0] = T[0] * Sf[0] + T[1] * Sf[3] + T[2] * Sf[6] + EPSF;
    C6[1] = T[0] * Sf[1] + T[1] * Sf[4] + T[2] * Sf[7];
    C6[2] = T[0] * Sf[2] + T[1] * Sf[5] + T[2] * Sf[8];
    C6[3] = T[3] * Sf[1] + T[4] * Sf[4] + T[5] * Sf[7] + EPSF;
    C6[4] = T[3] * Sf[2] + T[4] * Sf[5] + T[5] * Sf[8];
    C6[5] = T[6] * Sf[2] + T[7] * Sf[5] + T[8] * Sf[8] + EPSF;

    float CS6[6];
    sym_sqrt(C6, CS6);

    // V = A_sqrt * C_sqrt * A_sqrt_inv^T ; all three symmetric -> plain product
    float CSf[9], If[9], T2[9], V[9];
    sym_expand(CS6, CSf);
    sym_expand(I6, If);
    mm3(Sf, CSf, T2);
    mm3(T2, If, V);

    // cov_velocity = V + V^T - 2*cov_1 ; stage into LDS (reuse cov1 tile)
#pragma unroll
    for (int i = 0; i < 3; ++i) {
#pragma unroll
      for (int j = 0; j < 3; ++j) {
        lds_c1[tid * 9 + i * 3 + j] =
            V[i * 3 + j] + V[j * 3 + i] - 2.0f * c1[i * 3 + j];
      }
    }
  }

  __syncthreads();      // results visible block-wide before coalesced drain

  // ---- async coalesced LDS->global drain of cov_velocity ----
#pragma unroll
  for (int k = 0; k < 3; ++k) {
    int i4 = k * TPB + tid;
    if (i4 < (TPB * 9) / 4) {
      int gf = base_f + i4 * 4;
      if (gf + 4 <= nfTot) {
        async_store_b128(out_cov + gf, ldsBase1 + (unsigned)i4 * 16u);
      } else if (gf < nfTot) {
#pragma unroll
        for (int j = 0; j < 4; ++j) {
          if (gf + j < nfTot) {
            async_store_b32(out_cov + gf + j, ldsBase1 + (unsigned)(i4 * 16 + j * 4));
          }
        }
      }
    }
  }
  wait_async0();        // belt-and-braces; S_ENDPGM also implies wait-idle
}

// ---------- launch ----------

extern "C" void kernel_launch(void* const* d_in, const int* in_sizes, int n_in,
                              void* d_out, int out_size, void* d_ws, size_t ws_size,
                              hipStream_t stream) {
  const float* miu1 = (const float*)d_in[0];
  const float* miu2 = (const float*)d_in[1];
  const float* cov1 = (const float*)d_in[2];
  const float* cov2 = (const float*)d_in[3];
  const int B = in_sizes[0] / 3;

  float* out = (float*)d_out;
  float* out_miu = out;                 // first tuple output: (B,3)
  float* out_cov = out + (size_t)B * 3; // second tuple output: (B,3,3)

  const int blocks = (B + TPB - 1) / TPB;
  wasserstein_log_kernel<<<blocks, TPB, 0, stream>>>(
      miu1, miu2, cov1, cov2, out_miu, out_cov, B);
}